// edge_conv_60009283059881
// MI455X (gfx1250) — compile-verified
//
#include <hip/hip_runtime.h>

// EdgeConv, factored form:
//   P[m,:]  = X[m,:] @ W1^T                      (stored in d_ws, 80000x128 f32)
//   Q[m,:]  = X[m,:] @ (W2-W1)^T + (b1+b2)       (stored in d_out)
//   y[m,o]  = max_k max(0, Q[m,o] + P[row_k,o])  (in-place over d_out)
//
// GEMM phase uses V_WMMA_F32_16X16X4_F32 (wave32, 16x16 tiles, K-steps of 4).

typedef float v2f __attribute__((ext_vector_type(2)));
typedef float v4f __attribute__((ext_vector_type(4)));
typedef float v8f __attribute__((ext_vector_type(8)));

#define BATCH 4
#define NPTS  20000
#define KNN   16
#define CIN   64
#define COUT  128
#define MTOT  (BATCH * NPTS)   // 80000, divisible by 16 and by 8

// ---------------------------------------------------------------------------
// Phase A: per-point projections via WMMA f32 16x16x4.
// One wave handles a 16-row M-tile and the full OC=128 extent (8 N-tiles).
// A-matrix layout (ISA 7.12.2, 32-bit A 16x4): lane L<16 -> row M=L, holds
// {K0,K1}; lane L>=16 -> row M=L-16, holds {K2,K3}. So each lane's A frag for
// K-step kk is the contiguous float2  X[row, 4*kk + 2*hi .. +1].
// B-matrix 4x16: VGPR r holds row K=r (lanes 0-15) / K=r+2 (lanes 16-31),
// i.e. lane holds B[2*hi+{0,1}, col] = W[col, 4*kk + 2*hi + {0,1}] -> also a
// contiguous float2 from the row-major [OC,CIN] weights.
// ---------------------------------------------------------------------------
__global__ __launch_bounds__(256)
void edge_proj_wmma(const float* __restrict__ x,
                    const float* __restrict__ w1, const float* __restrict__ b1,
                    const float* __restrict__ w2, const float* __restrict__ b2,
                    float* __restrict__ P, float* __restrict__ Q)
{
    const int lane  = threadIdx.x & 31;
    const int wave  = threadIdx.x >> 5;
    const int mtile = blockIdx.x * 8 + wave;
    if (mtile * 16 >= MTOT) return;          // wave-uniform: EXEC stays all-1s

    const int hi  = lane >> 4;               // 0 or 1 (lane half)
    const int lo  = lane & 15;
    const int row = mtile * 16 + lo;

    // Preload all 16 A fragments for this M-tile (2 VGPRs each).
    v2f a[16];
    const float* xr = x + row * CIN;
    #pragma unroll
    for (int kk = 0; kk < 16; ++kk)
        a[kk] = *(const v2f*)(xr + kk * 4 + 2 * hi);

    for (int nt = 0; nt < 8; ++nt) {
        const int col = nt * 16 + lo;        // output channel (B/D column)
        const float bv = b1[col] + b2[col];  // fold both biases into Q

        v8f accP = {0.f, 0.f, 0.f, 0.f, 0.f, 0.f, 0.f, 0.f};
        v8f accQ = {bv, bv, bv, bv, bv, bv, bv, bv};

        const float* w1r = w1 + col * CIN;
        const float* w2r = w2 + col * CIN;
        #pragma unroll
        for (int kk = 0; kk < 16; ++kk) {
            v2f bw1 = *(const v2f*)(w1r + kk * 4 + 2 * hi);
            v2f bw2 = *(const v2f*)(w2r + kk * 4 + 2 * hi);
            v2f bwd = bw2 - bw1;             // (W2 - W1) fragment on the fly
            accP = __builtin_amdgcn_wmma_f32_16x16x4_f32(
                false, a[kk], false, bw1, (short)0, accP, false, false);
            accQ = __builtin_amdgcn_wmma_f32_16x16x4_f32(
                false, a[kk], false, bwd, (short)0, accQ, false, false);
        }

        // D layout: VGPR r -> row M = r (lanes 0-15) / r+8 (lanes 16-31).
        float* pr = P + (mtile * 16 + 8 * hi) * COUT + col;
        float* qr = Q + (mtile * 16 + 8 * hi) * COUT + col;
        #pragma unroll
        for (int r = 0; r < 8; ++r) {
            pr[r * COUT] = accP[r];
            qr[r * COUT] = accQ[r];
        }
    }
}

// ---------------------------------------------------------------------------
// Phase B: gather + relu + max over K neighbors.  One wave per point; each
// lane owns 4 channels -> every neighbor row read is one coalesced 512B
// global_load_b128 burst (P table is L2-resident: 41 MB << 192 MB).
// Q is read from d_out and the result written back in place (each lane reads
// then overwrites only its own 16 bytes).
// ---------------------------------------------------------------------------
__global__ __launch_bounds__(256)
void edge_gather_max(const float* __restrict__ P,
                     const int* __restrict__ idx,
                     float* __restrict__ QY)
{
    const int lane = threadIdx.x & 31;
    const int wave = threadIdx.x >> 5;
    const int m    = blockIdx.x * 8 + wave;
    if (m >= MTOT) return;

    const int b    = m / NPTS;
    const int base = b * NPTS;               // neighbor rows index within batch

    int iv = 0;
    if (lane < KNN) iv = idx[m * KNN + lane];

    const int c4 = lane * 4;                 // this lane's 4 channels
    v4f q   = *(const v4f*)(QY + m * COUT + c4);
    v4f acc = {0.f, 0.f, 0.f, 0.f};          // max_k relu(z) == max(0, max_k z)

    #pragma unroll
    for (int k = 0; k < KNN; ++k) {
        int r = __shfl(iv, k, 32);           // broadcast neighbor index (wave32)
        v4f p = *(const v4f*)(P + (size_t)(base + r) * COUT + c4);
        v4f z = q + p;
        acc.x = fmaxf(acc.x, z.x);
        acc.y = fmaxf(acc.y, z.y);
        acc.z = fmaxf(acc.z, z.z);
        acc.w = fmaxf(acc.w, z.w);
    }
    *(v4f*)(QY + m * COUT + c4) = acc;
}

extern "C" void kernel_launch(void* const* d_in, const int* in_sizes, int n_in,
                              void* d_out, int out_size, void* d_ws, size_t ws_size,
                              hipStream_t stream)
{
    const float* x   = (const float*)d_in[0];
    const int*   idx = (const int*)  d_in[1];   // JAX x64 disabled -> int32
    const float* w1  = (const float*)d_in[2];
    const float* b1  = (const float*)d_in[3];
    const float* w2  = (const float*)d_in[4];
    const float* b2  = (const float*)d_in[5];

    float* Q = (float*)d_out;                 // Q then overwritten with y
    float* P = (float*)d_ws;                  // needs MTOT*COUT*4 = 40.96 MB

    // Phase A: 80000/16 = 5000 M-tiles, 8 waves (tiles) per 256-thread block.
    edge_proj_wmma<<<dim3((MTOT / 16 + 7) / 8), dim3(256), 0, stream>>>(
        x, w1, b1, w2, b2, P, Q);

    // Phase B: 8 points per 256-thread block.
    edge_gather_max<<<dim3((MTOT + 7) / 8), dim3(256), 0, stream>>>(P, idx, Q);
}